// Attention_21217138442906
// MI455X (gfx1250) — compile-verified
//
#include <hip/hip_runtime.h>
#include <math.h>

// ---- problem constants (from reference) ----
#define NB 8
#define L0 128
#define L1 256
#define DD 512
#define NEG_BIG 1e12f

typedef float v2f __attribute__((ext_vector_type(2)));
typedef float v8f __attribute__((ext_vector_type(8)));

// pointer types demanded by the async-LDS builtin (per hipcc diagnostic):
// arg0: v4i in AS(1) (global, printed as __device__), arg1: v4i in AS(3) (LDS)
typedef int v4i __attribute__((vector_size(16)));
typedef __attribute__((address_space(1))) v4i glb_v4i;
typedef __attribute__((address_space(3))) v4i lds_v4i;

// ---- hardware tanh: the single most important instruction in this op ----
__device__ __forceinline__ float dtanhf(float x) {
#if __has_builtin(__builtin_amdgcn_tanhf)
  return __builtin_amdgcn_tanhf(x);            // CDNA5 v_tanh_f32
#else
  float r;
  asm("v_tanh_f32 %0, %1" : "=v"(r) : "v"(x)); // force the HW transcendental
  return r;
#endif
}

__device__ __forceinline__ v8f wmma_f32(v2f a, v2f b, v8f c) {
  // (neg_a, A, neg_b, B, c_mod, C, reuse_a, reuse_b)
  return __builtin_amdgcn_wmma_f32_16x16x4_f32(false, a, false, b, (short)0, c,
                                               false, false);
}

// ---- async global->LDS staging (gfx1250 path), safe fallback ----
__device__ __forceinline__ void stage_16xD(float* dst_lds, const float* src_glb,
                                           int tid) {
#if __has_builtin(__builtin_amdgcn_global_load_async_to_lds_b128)
  for (int i = tid; i < (16 * DD) / 4; i += 256)
    __builtin_amdgcn_global_load_async_to_lds_b128(
        (glb_v4i*)(src_glb + 4 * i), (lds_v4i*)(dst_lds + 4 * i), 0, 0);
#if __has_builtin(__builtin_amdgcn_s_wait_asynccnt)
  __builtin_amdgcn_s_wait_asynccnt(0);
#else
  asm volatile("s_wait_asynccnt 0" ::: "memory");
#endif
#else
  for (int i = tid; i < (16 * DD) / 4; i += 256)
    ((float4*)dst_lds)[i] = ((const float4*)src_glb)[i];
#endif
}

// -----------------------------------------------------------------------------
// Kernel 1: projections.  Rows [0,1024) = x -> XP (with +Wb fused),
// rows [1024,3072) = m -> MP.  One wave per 16x16 tile; K-loop of 128 WMMAs
// split over two accumulators for ILP.
// -----------------------------------------------------------------------------
__global__ __launch_bounds__(256)
void proj_kernel(const float* __restrict__ x, const float* __restrict__ m,
                 const float* __restrict__ Ww, const float* __restrict__ Wb,
                 float* __restrict__ XP, float* __restrict__ MP) {
  const int tid  = threadIdx.x;
  const int lid  = tid & 31;
  const int wv   = tid >> 5;
  const int ln   = lid & 15;
  const int koff = (lid >> 4) * 2;              // lanes 16-31 hold K=2,3

  const int rowbase = blockIdx.x * 16;          // 192 blocks
  const bool isx    = rowbase < (NB * L0);

  const float* src = isx ? (x + (size_t)rowbase * DD)
                         : (m + (size_t)(rowbase - NB * L0) * DD);
  float*       dst = isx ? (XP + (size_t)rowbase * DD)
                         : (MP + (size_t)(rowbase - NB * L0) * DD);
  const int coff = isx ? 0 : DD;                // Wx vs Wm columns of W_w

  const float* arow = src + (size_t)ln * DD;

  for (int i = 0; i < 4; ++i) {                 // 4 e-tiles per wave
    const int e0 = (wv * 4 + i) * 16;
    const float* bcol = Ww + (size_t)(e0 + ln) * (2 * DD) + coff;
    v8f acc0 = {}, acc1 = {};                   // break WMMA dep chain
#pragma unroll 2
    for (int d0 = 0; d0 < DD; d0 += 8) {
      v2f a0, b0, a1, b1;
      a0.x = arow[d0 + koff];     a0.y = arow[d0 + koff + 1];
      b0.x = bcol[d0 + koff];     b0.y = bcol[d0 + koff + 1];
      a1.x = arow[d0 + 4 + koff]; a1.y = arow[d0 + 4 + koff + 1];
      b1.x = bcol[d0 + 4 + koff]; b1.y = bcol[d0 + 4 + koff + 1];
      acc0 = wmma_f32(a0, b0, acc0);
      acc1 = wmma_f32(a1, b1, acc1);
    }
    const float bias = isx ? Wb[e0 + ln] : 0.0f;
    const int mb = (lid >> 4) * 8;
#pragma unroll
    for (int r = 0; r < 8; ++r)
      dst[(size_t)(mb + r) * DD + e0 + ln] = acc0[r] + acc1[r] + bias;
  }
}

// -----------------------------------------------------------------------------
// Kernel 2: fused tanh-logits + masked softmax + v = w @ m.
// One block = one (n, 16-row a-tile).  LDS: xp tile (32KB) | mp chunk (32KB),
// front half reused for softmax weights before the WMMA stage.
// -----------------------------------------------------------------------------
__global__ __launch_bounds__(256)
void attn_kernel(const float* __restrict__ m, const int* __restrict__ mask,
                 const float* __restrict__ Vw, const float* __restrict__ Vb,
                 const float* __restrict__ XP, const float* __restrict__ MP,
                 float* __restrict__ out) {
  __shared__ float smem[2 * 16 * DD];            // 64 KB
  float* s_xp = smem;
  float* s_mp = smem + 16 * DD;

  const int tid = threadIdx.x;
  const int lid = tid & 31;
  const int wv  = tid >> 5;

  const int n  = blockIdx.x >> 3;
  const int a0 = (blockIdx.x & 7) * 16;

  // stage XP tile (already contains W_b)
  stage_16xD(s_xp, XP + (size_t)(n * L0 + a0) * DD, tid);

  const float vb = Vb[0];
  const int la = tid >> 4;                       // this thread's a-row (0..15)
  const int lb = tid & 15;                       // b sub-lane
  const float* mp_g = MP + (size_t)(n * L1) * DD;
  const float4* vw4 = (const float4*)Vw;         // wave-uniform -> s_load_b128

  float lg[16];
  for (int ch = 0; ch < 16; ++ch) {
    __syncthreads();                             // s_mp reuse / s_xp ready
    stage_16xD(s_mp, mp_g + (size_t)(ch * 16) * DD, tid);
    __syncthreads();

    // hot loop: ds_load_b128 x2 + 4x (v_add, v_tanh_f32, v_fma) per iter
    float acc = 0.f;
    const float4* xr = (const float4*)(s_xp + la * DD);
    const float4* mr = (const float4*)(s_mp + lb * DD);
#pragma unroll 4
    for (int d = 0; d < DD / 4; ++d) {
      const float4 xv = xr[d], mv = mr[d], vv = vw4[d];
      acc += dtanhf(xv.x + mv.x) * vv.x;
      acc += dtanhf(xv.y + mv.y) * vv.y;
      acc += dtanhf(xv.z + mv.z) * vv.z;
      acc += dtanhf(xv.w + mv.w) * vv.w;
    }
    const int b  = lb + ch * 16;
    lg[ch] = (mask[n * L1 + b] == 0) ? -NEG_BIG : (acc + vb);
  }

  // row softmax: 16 regs/thread x 16 lanes per row; wave32 xor-shuffles 1..8
  float mx = -3.0e38f;
#pragma unroll
  for (int ch = 0; ch < 16; ++ch) mx = fmaxf(mx, lg[ch]);
#pragma unroll
  for (int s = 1; s < 16; s <<= 1) mx = fmaxf(mx, __shfl_xor(mx, s, 32));
  float sum = 0.f;
#pragma unroll
  for (int ch = 0; ch < 16; ++ch) { lg[ch] = __expf(lg[ch] - mx); sum += lg[ch]; }
#pragma unroll
  for (int s = 1; s < 16; s <<= 1) sum += __shfl_xor(sum, s, 32);
  const float inv = 1.0f / sum;

  __syncthreads();                               // done with s_xp reads
  float* s_w = smem;                             // reuse front 16KB: 16 x 256
#pragma unroll
  for (int ch = 0; ch < 16; ++ch)
    s_w[la * L1 + lb + ch * 16] = lg[ch] * inv;
  __syncthreads();

  // v = w (16x256, LDS) @ m (256x512, L2-resident), f32 WMMA, 2 accumulators
  const int ln   = lid & 15;
  const int koff = (lid >> 4) * 2;
  const float* mrow = m + (size_t)(n * L1) * DD;
#pragma unroll
  for (int i = 0; i < 4; ++i) {                  // 4 d-tiles per wave
    const int d0 = (wv * 4 + i) * 16;
    v8f acc0 = {}, acc1 = {};
    for (int b0 = 0; b0 < L1; b0 += 8) {
      v2f a0, b0f, a1, b1f;
      a0.x  = s_w[ln * L1 + b0 + koff];
      a0.y  = s_w[ln * L1 + b0 + koff + 1];
      b0f.x = mrow[(size_t)(b0 + koff) * DD + d0 + ln];
      b0f.y = mrow[(size_t)(b0 + koff + 1) * DD + d0 + ln];
      a1.x  = s_w[ln * L1 + b0 + 4 + koff];
      a1.y  = s_w[ln * L1 + b0 + 4 + koff + 1];
      b1f.x = mrow[(size_t)(b0 + 4 + koff) * DD + d0 + ln];
      b1f.y = mrow[(size_t)(b0 + 4 + koff + 1) * DD + d0 + ln];
      acc0 = wmma_f32(a0, b0f, acc0);
      acc1 = wmma_f32(a1, b1f, acc1);
    }
    const int mb = (lid >> 4) * 8;
#pragma unroll
    for (int r = 0; r < 8; ++r)
      out[(size_t)(n * L0 + a0 + mb + r) * DD + d0 + ln] = acc0[r] + acc1[r];
  }
}

// -----------------------------------------------------------------------------
extern "C" void kernel_launch(void* const* d_in, const int* in_sizes, int n_in,
                              void* d_out, int out_size, void* d_ws, size_t ws_size,
                              hipStream_t stream) {
  const float* x    = (const float*)d_in[0];
  const float* m    = (const float*)d_in[1];
  const int*   mask = (const int*)d_in[2];
  const float* Ww   = (const float*)d_in[3];
  const float* Wb   = (const float*)d_in[4];
  const float* Vw   = (const float*)d_in[5];
  const float* Vb   = (const float*)d_in[6];
  float* out = (float*)d_out;

  float* XP = (float*)d_ws;                          // 1024*512 f32 = 2 MB
  float* MP = XP + (size_t)NB * L0 * DD;             // 2048*512 f32 = 4 MB

  proj_kernel<<<192, 256, 0, stream>>>(x, m, Ww, Wb, XP, MP);
  attn_kernel<<<64, 256, 0, stream>>>(m, mask, Vw, Vb, XP, MP, out);
}